// IntermediateKeypointPredictor_29145648070738
// MI455X (gfx1250) — compile-verified
//
#include <hip/hip_runtime.h>
#include <math.h>

typedef __attribute__((ext_vector_type(2))) float v2f;
typedef __attribute__((ext_vector_type(8))) float v8f;

#define BB   16
#define CC   64
#define HH   96
#define WW   96
#define HWN  (HH*WW)          // 9216
#define NAA  20
#define NII  50
#define PP   (NAA*NII)        // 1000
#define T32  32               // ceil(1000/32) tiles of 32 P-rows
#define NWAVES 4              // HW split across waves in a workgroup
#define JTOT (HWN/16)         // 576 column chunks
#define JPW  (JTOT/NWAVES)    // 144 chunks per wave
// Fold the softmax multiplier AND the exp->exp2 base change into A:
// softmax_e(50*c) == softmax_2((50*log2(e))*c), so pre-scale descriptors once.
#define ASCALE (50.0f * 1.44269504088896340736f)
#define CSTEP  (2.0f / 95.0f)   // linspace(-1,1,96) step

// Native base-2 exp: v_exp_f32
__device__ __forceinline__ float fast_exp2(float x) {
  return __builtin_amdgcn_exp2f(x);
}

// Workgroup = 4 waves, all on the same 32-row P tile, each owning a quarter of HW.
// Per wave: two resident fp32 A tiles (ISA 16x4 layout), register-double-buffered
// B tiles, 32 chained V_WMMA_F32_16X16X4_F32 per chunk, base-2 online softmax in
// VGPRs, wave-local shfl_xor butterfly, then cross-wave merge through LDS.
__global__ __launch_bounds__(32*NWAVES)
void corr_softargmax_kernel(const float* __restrict__ x,
                            const float* __restrict__ desc,
                            float* __restrict__ out)
{
  __shared__ float lds[32][NWAVES][4];   // [row-in-tile][wave][m,se,sx,sy]

  const int b    = blockIdx.x / T32;
  const int t    = blockIdx.x % T32;
  const int tid  = threadIdx.x;
  const int wave = tid >> 5;
  const int lane = tid & 31;
  const int half = lane >> 4;            // 0: lanes 0-15, 1: lanes 16-31
  const int nl   = lane & 15;
  const int p0   = t * 32;

  // ---- Resident A tiles: A[M][K] = ASCALE * descriptors[K][p], rows p0..p0+31 ----
  v2f aLo[16], aHi[16];
  {
    const int  pl  = p0 + nl;        const bool okl = (pl < PP);
    const int  ph  = p0 + 16 + nl;   const bool okh = (ph < PP);
    #pragma unroll
    for (int kc = 0; kc < 16; ++kc) {
      const int k0 = 4 * kc + 2 * half;
      aLo[kc].x = okl ? ASCALE * desc[(size_t)(k0    ) * PP + pl] : 0.0f;
      aLo[kc].y = okl ? ASCALE * desc[(size_t)(k0 + 1) * PP + pl] : 0.0f;
      aHi[kc].x = okh ? ASCALE * desc[(size_t)(k0    ) * PP + ph] : 0.0f;
      aHi[kc].y = okh ? ASCALE * desc[(size_t)(k0 + 1) * PP + ph] : 0.0f;
    }
  }

  // ---- Online softmax state: 16 rows per lane (r<8 -> lo tile, r>=8 -> hi) ----
  float m[16], se[16], sx[16], sy[16];
  #pragma unroll
  for (int r = 0; r < 16; ++r) { m[r] = -1e30f; se[r] = 0.f; sx[r] = 0.f; sy[r] = 0.f; }

  const float* xb = x + (size_t)b * CC * HWN;
  const int jbeg = wave * JPW;
  const int jend = jbeg + JPW;

  v2f bv0[16], bv1[16];

  auto loadB = [&](v2f (&bv)[16], int j) {
    const int n = j * 16 + nl;
    #pragma unroll
    for (int kc = 0; kc < 16; ++kc) {
      const int k0 = 4 * kc + 2 * half;
      bv[kc].x = xb[(size_t)(k0    ) * HWN + n];
      bv[kc].y = xb[(size_t)(k0 + 1) * HWN + n];
    }
  };

  // jm6 = j % 6, jd6 = j / 6  (96 = 6*16, so col = jm6*16+nl, row = jd6)
  auto compute = [&](v2f (&bv)[16], int jm6, int jd6) {
    v8f accLo = {}, accHi = {};
    #pragma unroll
    for (int kc = 0; kc < 16; ++kc)
      accLo = __builtin_amdgcn_wmma_f32_16x16x4_f32(false, aLo[kc], false, bv[kc],
                                                    (short)0, accLo, false, false);
    #pragma unroll
    for (int kc = 0; kc < 16; ++kc)
      accHi = __builtin_amdgcn_wmma_f32_16x16x4_f32(false, aHi[kc], false, bv[kc],
                                                    (short)0, accHi, false, false);
    const float xx = (float)(jm6 * 16 + nl) * CSTEP - 1.0f;
    const float yy = (float)(jd6)           * CSTEP - 1.0f;
    #pragma unroll
    for (int r = 0; r < 16; ++r) {
      const float c  = (r < 8) ? accLo[r] : accHi[r - 8];   // already *50*log2e
      const float nm = fmaxf(m[r], c);
      const float sc = fast_exp2(m[r] - nm);
      const float e  = fast_exp2(c - nm);
      se[r] = se[r] * sc + e;
      sx[r] = sx[r] * sc + e * xx;
      sy[r] = sy[r] * sc + e * yy;
      m[r]  = nm;
    }
  };

  // ---- Main loop, register double-buffered B, incremental col/row counters ----
  int jm6 = 0;            // jbeg = wave*144, 144 % 6 == 0
  int jd6 = wave * 24;    // (wave*144*16)/96
  loadB(bv0, jbeg);
  for (int j = jbeg; j < jend; j += 2) {
    loadB(bv1, j + 1);
    compute(bv0, jm6, jd6);
    jm6++; if (jm6 == 6) { jm6 = 0; jd6++; }
    const int jn = (j + 2 < jend) ? (j + 2) : jbeg;   // harmless wrap on last iter
    loadB(bv0, jn);
    compute(bv1, jm6, jd6);
    jm6++; if (jm6 == 6) { jm6 = 0; jd6++; }
  }

  // ---- Wave-local butterfly merge over the 16 lanes of each half ----
  #pragma unroll
  for (int mask = 1; mask <= 8; mask <<= 1) {
    #pragma unroll
    for (int r = 0; r < 16; ++r) {
      const float mo = __shfl_xor(m[r],  mask, 32);
      const float so = __shfl_xor(se[r], mask, 32);
      const float xo = __shfl_xor(sx[r], mask, 32);
      const float yo = __shfl_xor(sy[r], mask, 32);
      const float nm = fmaxf(m[r], mo);
      const float s1 = fast_exp2(m[r] - nm);
      const float s2 = fast_exp2(mo   - nm);
      se[r] = se[r] * s1 + so * s2;
      sx[r] = sx[r] * s1 + xo * s2;
      sy[r] = sy[r] * s1 + yo * s2;
      m[r]  = nm;
    }
  }

  // ---- Publish per-wave partials to LDS ----
  if (nl == 0) {
    #pragma unroll
    for (int r = 0; r < 16; ++r) {
      const int row = (r < 8) ? (r + 8 * half) : (16 + (r - 8) + 8 * half);
      lds[row][wave][0] = m[r];
      lds[row][wave][1] = se[r];
      lds[row][wave][2] = sx[r];
      lds[row][wave][3] = sy[r];
    }
  }
  __syncthreads();

  // ---- Cross-wave merge + final write: one thread per tile row ----
  if (tid < 32) {
    const int row = tid;
    const int p   = p0 + row;
    if (p < PP) {
      float M = lds[row][0][0], S = lds[row][0][1];
      float X = lds[row][0][2], Y = lds[row][0][3];
      #pragma unroll
      for (int w = 1; w < NWAVES; ++w) {
        const float mo = lds[row][w][0], so = lds[row][w][1];
        const float xo = lds[row][w][2], yo = lds[row][w][3];
        const float nm = fmaxf(M, mo);
        const float s1 = fast_exp2(M  - nm);
        const float s2 = fast_exp2(mo - nm);
        S = S * s1 + so * s2;
        X = X * s1 + xo * s2;
        Y = Y * s1 + yo * s2;
        M = nm;
      }
      const size_t base = 640 + 2 * ((size_t)b * PP + p);
      out[base + 0] = X / S;
      out[base + 1] = Y / S;
    }
  }
}

// pred[b,na,k] = sum_i intermediate[b,na,i] * w_lin[k,i], i over nI*2 = 100
__global__ void pred_kernel(const float* __restrict__ w_lin, float* __restrict__ out)
{
  const int tid = blockIdx.x * blockDim.x + threadIdx.x;
  if (tid >= BB * NAA * 2) return;
  const int k  = tid & 1;
  const int bn = tid >> 1;                       // b*NA + na
  const float* inter = out + 640 + (size_t)bn * (NII * 2);
  const float* wr    = w_lin + k * (NII * 2);
  float s = 0.0f;
  #pragma unroll 4
  for (int i = 0; i < NII * 2; ++i) s += inter[i] * wr[i];
  out[tid] = s;
}

extern "C" void kernel_launch(void* const* d_in, const int* in_sizes, int n_in,
                              void* d_out, int out_size, void* d_ws, size_t ws_size,
                              hipStream_t stream)
{
  const float* x     = (const float*)d_in[0];   // (16,64,96,96)
  const float* desc  = (const float*)d_in[1];   // (64,20,50)
  const float* w_lin = (const float*)d_in[2];   // (2,100)
  float* out = (float*)d_out;                   // 640 pred + 32000 intermediate

  corr_softargmax_kernel<<<BB * T32, 32 * NWAVES, 0, stream>>>(x, desc, out);
  pred_kernel<<<(BB * NAA * 2 + 127) / 128, 128, 0, stream>>>(w_lin, out);
}